// _msg_nn_27230092657674
// MI455X (gfx1250) — compile-verified
//
#include <hip/hip_runtime.h>

typedef __attribute__((ext_vector_type(2))) float v2f;
typedef __attribute__((ext_vector_type(4))) float v4f;
typedef __attribute__((ext_vector_type(8))) float v8f;

#define B_DIM 16
#define N_DIM 32
#define H_DIM 64

// One block = 256 threads = 8 wave32s, handling one (b, j) pair.
// Wave w: h-block (w & 3) -> h0 = 16*(w&3); i-half (w >> 2) -> i in [16*ih, 16*ih+16).
// Each wave computes acc[0] lanes 0..15 = partial out[b, j, h0..h0+15]; LDS-reduce the
// two i-halves, then 64 threads store out[b,j,:].
__global__ __launch_bounds__(256)
void msg_nn_wmma(const float* __restrict__ W, const float* __restrict__ X,
                 float* __restrict__ out) {
  const int bj    = blockIdx.x;       // 0 .. B*N-1
  const int b     = bj >> 5;          // / N
  const int j     = bj & 31;          // % N
  const int wave  = threadIdx.x >> 5;
  const int lane  = threadIdx.x & 31;
  const int m     = lane & 15;        // column / row-in-half
  const int q     = lane >> 4;        // lane half (selects k-group)
  const int h0    = (wave & 3) << 4;  // h-block
  const int i0    = (wave >> 2) << 4; // i-half start

  // W element index: (((b*N + i)*N + j)*H + h)*H + g
  const size_t w_i_stride = (size_t)N_DIM * H_DIM * H_DIM;   // stride per i (floats)
  const float* wbase = W
      + ((size_t)((b * N_DIM + i0) * N_DIM + j)) * (H_DIM * H_DIM)
      + (size_t)(h0 + m) * H_DIM + 4 * q;
  const float* xbase = X + (size_t)(b * N_DIM + i0) * H_DIM + 4 * q;

  v8f acc = {0.f, 0.f, 0.f, 0.f, 0.f, 0.f, 0.f, 0.f};

  for (int i = 0; i < 16; ++i) {
    const float* wi = wbase + (size_t)i * w_i_stride;
    const float* xi = xbase + i * H_DIM;
    if (i < 15) __builtin_prefetch(wi + w_i_stride, 0, 0);  // next i slice -> global_prefetch_b8

#pragma unroll
    for (int G = 0; G < H_DIM; G += 16) {
      // Lane L holds g in [G+4q, G+4q+4) and [G+8+4q, G+8+4q+4): stream W non-temporally.
      v4f w0 = __builtin_nontemporal_load((const v4f*)(wi + G));
      v4f w1 = __builtin_nontemporal_load((const v4f*)(wi + G + 8));
      v4f x0 = *(const v4f*)(xi + G);       // tiny, cache-resident
      v4f x1 = *(const v4f*)(xi + G + 8);

      // 4 WMMAs per 16 g-values; g<->k mapping identical for A (x) and B (W).
      v2f a0 = {x0[0], x0[1]}, b0 = {w0[0], w0[1]};
      acc = __builtin_amdgcn_wmma_f32_16x16x4_f32(false, a0, false, b0, (short)0, acc, false, false);
      v2f a1 = {x0[2], x0[3]}, b1 = {w0[2], w0[3]};
      acc = __builtin_amdgcn_wmma_f32_16x16x4_f32(false, a1, false, b1, (short)0, acc, false, false);
      v2f a2 = {x1[0], x1[1]}, b2 = {w1[0], w1[1]};
      acc = __builtin_amdgcn_wmma_f32_16x16x4_f32(false, a2, false, b2, (short)0, acc, false, false);
      v2f a3 = {x1[2], x1[3]}, b3 = {w1[2], w1[3]};
      acc = __builtin_amdgcn_wmma_f32_16x16x4_f32(false, a3, false, b3, (short)0, acc, false, false);
    }
  }

  // acc[0], lanes 0..15: partial out for h = h0 + m (all C rows identical by construction).
  __shared__ float red[8 * 16];
  if (q == 0) red[wave * 16 + m] = acc[0];
  __syncthreads();

  if (threadIdx.x < 64) {
    const int hblk = threadIdx.x >> 4;
    const int mm   = threadIdx.x & 15;
    const float v  = red[hblk * 16 + mm] + red[(hblk + 4) * 16 + mm];
    out[((size_t)(b * N_DIM + j)) * H_DIM + threadIdx.x] = v * (1.0f / (float)N_DIM);
  }
}

extern "C" void kernel_launch(void* const* d_in, const int* in_sizes, int n_in,
                              void* d_out, int out_size, void* d_ws, size_t ws_size,
                              hipStream_t stream) {
  const float* W  = (const float*)d_in[0];  // edge_wgt [B,N,N,H,H] fp32
  const float* X  = (const float*)d_in[1];  // node_hidden [B,N,H] fp32
  float* out      = (float*)d_out;          // [B,N,H] fp32

  dim3 grid(B_DIM * N_DIM);  // 512 blocks, one per (b, j)
  dim3 block(256);           // 8 wave32s
  hipLaunchKernelGGL(msg_nn_wmma, grid, block, 0, stream, W, X, out);
}